// DynamicMDETR_58282706206954
// MI455X (gfx1250) — compile-verified
//
#include <hip/hip_runtime.h>

typedef __attribute__((ext_vector_type(2))) float v2f;
typedef __attribute__((ext_vector_type(8))) float v8f;

#define C_DIM 256
#define NPTS  36
#define NOFF  72      // IN_POINTS*2
#define MAP_H 40
#define MAP_W 40
#define HW    1600

// --------------------------------------------------------------------------
// Kernel 1: xy_offsets = content_query @ W_off^T  (+ b_off)
// One wave per 16x16 output tile, V_WMMA_F32_16X16X4_F32, K-loop of 4.
// Column/row masking is done at the STORE only: garbage B columns (n>=72)
// and clamped A rows only affect D elements that are never written.
// --------------------------------------------------------------------------
__global__ __launch_bounds__(32)
void offsets_wmma_kernel(const float* __restrict__ cq,
                         const float* __restrict__ Woff,
                         const float* __restrict__ boff,
                         float* __restrict__ off_out, int bs)
{
    const int lane = threadIdx.x & 31;
    const int half = lane >> 4;      // 0: lanes 0-15, 1: lanes 16-31
    const int lrow = lane & 15;
    const int m0 = blockIdx.x * 16;
    const int n0 = blockIdx.y * 16;
    const int n  = n0 + lrow;
    const bool nvalid = (n < NOFF);

    const int mrow = min(m0 + lrow, bs - 1);                        // addr-safe clamp
    const float* arow = cq + (size_t)mrow * C_DIM;                  // A row
    const float* brow = Woff + (size_t)(nvalid ? n : 0) * C_DIM;    // B col n = W_off row n

    v8f acc = {0.f, 0.f, 0.f, 0.f, 0.f, 0.f, 0.f, 0.f};
    for (int k0 = 0; k0 < C_DIM; k0 += 4) {
        const int k = k0 + 2 * half;   // A: v0=K(2h), v1=K(2h+1); B rows likewise
        v2f a, b;
        a.x = arow[k];
        a.y = arow[k + 1];
        b.x = brow[k];
        b.y = brow[k + 1];
        acc = __builtin_amdgcn_wmma_f32_16x16x4_f32(
            /*neg_a=*/false, a, /*neg_b=*/false, b,
            /*c_mod=*/(short)0, acc, /*reuse_a=*/false, /*reuse_b=*/false);
    }

    if (nvalid) {
        const float bias = boff[n];
        #pragma unroll
        for (int r = 0; r < 8; ++r) {
            const int m = m0 + r + 8 * half;   // C/D: VGPR r -> M=r (+8 for hi half)
            if (m < bs) off_out[(size_t)m * NOFF + n] = acc[r] + bias;
        }
    }
}

// --------------------------------------------------------------------------
// Kernel 2: bilinear grid-sample (border padding, align_corners=False) of
// feature_map and pos at 36 points/batch. Block = batch, thread = channel.
// Point coordinates depend only on (b,p): threads 0..35 compute the four
// gather indices + weights once into LDS; all 256 channel threads then do
// pure gather+FMA (ds_load_b128 per point, per-thread contiguous stores).
// --------------------------------------------------------------------------
__global__ __launch_bounds__(C_DIM)
void sample_kernel(const float* __restrict__ fmap,
                   const float* __restrict__ pos,
                   const float* __restrict__ ref,
                   const float* __restrict__ off,
                   float* __restrict__ outF,
                   float* __restrict__ outP)
{
    __shared__ int4   s_idx[NPTS];
    __shared__ float4 s_wgt[NPTS];

    const int b = blockIdx.x;
    const int c = threadIdx.x;

    if (c < NPTS) {
        const int p = c;
        const float sx = off[b * NOFF + 2 * p + 0] + ref[b * 2 + 0];
        const float sy = off[b * NOFF + 2 * p + 1] + ref[b * 2 + 1];

        // grid = 2s-1; unnormalize (align_corners=False) -> pix = size*s - 0.5
        const float x = fminf(fmaxf((float)MAP_W * sx - 0.5f, 0.f), (float)(MAP_W - 1));
        const float y = fminf(fmaxf((float)MAP_H * sy - 0.5f, 0.f), (float)(MAP_H - 1));
        const float x0f = floorf(x), y0f = floorf(y);
        const float wx = x - x0f,   wy = y - y0f;
        const int x0 = (int)x0f,    y0 = (int)y0f;
        const int x1 = min(x0 + 1, MAP_W - 1);
        const int y1 = min(y0 + 1, MAP_H - 1);

        s_idx[p] = make_int4(y0 * MAP_W + x0, y0 * MAP_W + x1,
                             y1 * MAP_W + x0, y1 * MAP_W + x1);
        s_wgt[p] = make_float4((1.f - wx) * (1.f - wy), wx * (1.f - wy),
                               (1.f - wx) * wy,         wx * wy);
    }
    __syncthreads();

    const float* fm = fmap + ((size_t)b * C_DIM + c) * HW;
    const float* pm = pos  + ((size_t)b * C_DIM + c) * HW;
    float* oF = outF + ((size_t)b * C_DIM + c) * NPTS;
    float* oP = outP + ((size_t)b * C_DIM + c) * NPTS;

    __builtin_prefetch(fm, 0, 1);   // global_prefetch_b8: pull channel rows
    __builtin_prefetch(pm, 0, 1);

    for (int p = 0; p < NPTS; ++p) {
        const int4   ii = s_idx[p];
        const float4 ww = s_wgt[p];
        oF[p] = fm[ii.x] * ww.x + fm[ii.y] * ww.y + fm[ii.z] * ww.z + fm[ii.w] * ww.w;
        oP[p] = pm[ii.x] * ww.x + pm[ii.y] * ww.y + pm[ii.z] * ww.z + pm[ii.w] * ww.w;
    }
}

// --------------------------------------------------------------------------
extern "C" void kernel_launch(void* const* d_in, const int* in_sizes, int n_in,
                              void* d_out, int out_size, void* d_ws, size_t ws_size,
                              hipStream_t stream)
{
    const float* cq   = (const float*)d_in[0];  // content_query  (bs, 256)
    const float* ref  = (const float*)d_in[1];  // reference_point (bs, 2)
    const float* fmap = (const float*)d_in[2];  // feature_map (bs, 256, 1600)
    const float* pos  = (const float*)d_in[3];  // pos         (bs, 256, 1600)
    const float* Woff = (const float*)d_in[4];  // W_off (72, 256)
    const float* boff = (const float*)d_in[5];  // b_off (72)

    float* out = (float*)d_out;
    float* ws  = (float*)d_ws;                  // bs*72 floats of scratch offsets

    const int bs = in_sizes[1] / 2;             // reference_point is (bs, 2)

    dim3 g1((bs + 15) / 16, (NOFF + 15) / 16);  // 8 x 5 tiles, 1 wave each
    offsets_wmma_kernel<<<g1, 32, 0, stream>>>(cq, Woff, boff, ws, bs);

    float* outF = out;
    float* outP = out + (size_t)bs * C_DIM * NPTS;
    sample_kernel<<<bs, C_DIM, 0, stream>>>(fmap, pos, ref, ws, outF, outP);
}